// message_75685913690747
// MI455X (gfx1250) — compile-verified
//
#include <hip/hip_runtime.h>
#include <math.h>

typedef __attribute__((ext_vector_type(2))) float v2f;
typedef __attribute__((ext_vector_type(8))) float v8f;

#define F        128
#define F3       384
#define NNODES   50000
#define NEDGES   200000
#define NRBF     20
#define CUTOFF_F 5.0f
#define PI_F     3.14159265358979323846f

__device__ __forceinline__ float silu_f(float x) {
    return x / (1.0f + __expf(-x));
}

// ---------------------------------------------------------------------------
// Kernel 1: phi = silu(s @ W1 + b1) @ W2 + b2   via V_WMMA_F32_16X16X4_F32
// One block (8 waves) per 16-node tile.
// ---------------------------------------------------------------------------
__global__ __launch_bounds__(256) void node_mlp_kernel(
    const float* __restrict__ s, const float* __restrict__ W1,
    const float* __restrict__ b1, const float* __restrict__ W2,
    const float* __restrict__ b2, float* __restrict__ phi)
{
    __shared__ float lds_h[16 * 130];   // padded stride 130 -> conflict-free A reads

    const int tile = blockIdx.x;
    const int tid  = threadIdx.x;
    const int wave = tid >> 5;
    const int lane = tid & 31;
    const int row0 = tile * 16;

    // ISA A-layout (16x4 fp32): lanes 0-15 hold K=0,1; lanes 16-31 hold K=2,3
    const int mrow  = lane & 15;
    const int koff  = (lane >> 4) * 2;
    // ISA C/D-layout: reg r, lane<16 -> M=r ; lane>=16 -> M=r+8 ; N = lane&15
    const int chalf = (lane >> 4) * 8;

    // ---- preload A (s tile, all K=128) into registers: 32 k-steps x float2
    v2f a[32];
    {
        const float* srow = s + (size_t)(row0 + mrow) * F;
        #pragma unroll
        for (int kk = 0; kk < 32; ++kk) {
            const int k = kk * 4 + koff;
            a[kk].x = srow[k];
            a[kk].y = srow[k + 1];
        }
    }

    // ---- stage 1: h = silu(s @ W1 + b1); this wave owns N-tile == wave
    {
        const int ncol = wave * 16 + mrow;        // B/D column for this lane
        v8f c = {};
        #pragma unroll
        for (int kk = 0; kk < 32; ++kk) {
            const int k = kk * 4 + koff;
            v2f b;
            b.x = W1[(size_t)(k    ) * F + ncol];
            b.y = W1[(size_t)(k + 1) * F + ncol];
            c = __builtin_amdgcn_wmma_f32_16x16x4_f32(
                    false, a[kk], false, b, (short)0, c, false, false);
        }
        const float bias = b1[ncol];
        #pragma unroll
        for (int r = 0; r < 8; ++r) {
            const int m = r + chalf;
            lds_h[m * 130 + ncol] = silu_f(c[r] + bias);
        }
    }
    __syncthreads();

    // ---- reload A = h tile from LDS (transposed-consistent with A layout)
    #pragma unroll
    for (int kk = 0; kk < 32; ++kk) {
        const int k = kk * 4 + koff;
        a[kk].x = lds_h[mrow * 130 + k];
        a[kk].y = lds_h[mrow * 130 + k + 1];
    }

    // ---- stage 2: phi = h @ W2 + b2 (16 x 384); wave owns n-tiles {wave, wave+8, wave+16}
    float* prow = phi + (size_t)row0 * F3;
    #pragma unroll
    for (int t = 0; t < 3; ++t) {
        const int ncol = (wave + t * 8) * 16 + mrow;
        v8f c = {};
        #pragma unroll
        for (int kk = 0; kk < 32; ++kk) {
            const int k = kk * 4 + koff;
            v2f b;
            b.x = W2[(size_t)(k    ) * F3 + ncol];
            b.y = W2[(size_t)(k + 1) * F3 + ncol];
            c = __builtin_amdgcn_wmma_f32_16x16x4_f32(
                    false, a[kk], false, b, (short)0, c, false, false);
        }
        const float bias = b2[ncol];
        #pragma unroll
        for (int r = 0; r < 8; ++r) {
            const int m = r + chalf;
            prow[(size_t)m * F3 + ncol] = c[r] + bias;
        }
    }
}

// ---------------------------------------------------------------------------
// Kernel 2: per-edge rbf projection + gather + scatter-add (memory bound).
// One wave per edge; lane owns features {lane, lane+32, lane+64, lane+96}.
// ---------------------------------------------------------------------------
__global__ __launch_bounds__(256) void edge_kernel(
    const float* __restrict__ v,   const float* __restrict__ rel_pos,
    const float* __restrict__ Wr,  const float* __restrict__ br,
    const int*   __restrict__ src, const int*   __restrict__ dst,
    const float* __restrict__ phi,
    float* __restrict__ out_v, float* __restrict__ out_s)
{
    const int wave = threadIdx.x >> 5;
    const int lane = threadIdx.x & 31;
    const int e = blockIdx.x * 8 + wave;
    if (e >= NEDGES) return;

    const float rx = rel_pos[e * 3 + 0];
    const float ry = rel_pos[e * 3 + 1];
    const float rz = rel_pos[e * 3 + 2];
    const float d  = sqrtf(rx * rx + ry * ry + rz * rz);
    const float inv_d = 1.0f / d;
    const float fcut = (d < CUTOFF_F)
                     ? 0.5f * (__cosf(PI_F * d * (1.0f / CUTOFF_F)) + 1.0f)
                     : 0.0f;

    float rbf[NRBF];
    #pragma unroll
    for (int n = 0; n < NRBF; ++n)
        rbf[n] = __sinf((float)(n + 1) * (PI_F / CUTOFF_F) * d) * inv_d;

    const int si = src[e];
    const int di = dst[e];
    const float dirx = rx * inv_d, diry = ry * inv_d, dirz = rz * inv_d;

    const float* phis = phi + (size_t)si * F3;
    const float* vs   = v   + (size_t)si * F3;   // v is [N,3,F] row-major
    float* outs = out_s + (size_t)di * F;
    float* outv = out_v + (size_t)di * F3;

    #pragma unroll
    for (int j = 0; j < 4; ++j) {
        const int f = lane + j * 32;
        float w0 = br[f], w1 = br[f + F], w2 = br[f + 2 * F];
        #pragma unroll
        for (int n = 0; n < NRBF; ++n) {
            const float r = rbf[n];
            const float* wrow = Wr + (size_t)n * F3 + f;
            w0 = fmaf(r, wrow[0],     w0);
            w1 = fmaf(r, wrow[F],     w1);
            w2 = fmaf(r, wrow[2 * F], w2);
        }
        const float sv = w0 * fcut * phis[f];           // split[:, :F]
        const float ss = w1 * fcut * phis[f + F];       // split[:, F:2F]
        const float sr = w2 * fcut * phis[f + 2 * F];   // split[:, 2F:]

        atomicAdd(outs + f, ss);
        atomicAdd(outv + 0 * F + f, fmaf(vs[0 * F + f], sv, dirx * sr));
        atomicAdd(outv + 1 * F + f, fmaf(vs[1 * F + f], sv, diry * sr));
        atomicAdd(outv + 2 * F + f, fmaf(vs[2 * F + f], sv, dirz * sr));
    }
}

// ---------------------------------------------------------------------------
extern "C" void kernel_launch(void* const* d_in, const int* in_sizes, int n_in,
                              void* d_out, int out_size, void* d_ws, size_t ws_size,
                              hipStream_t stream) {
    const float* s   = (const float*)d_in[0];
    const float* v   = (const float*)d_in[1];
    const float* rel = (const float*)d_in[2];
    const float* W1  = (const float*)d_in[3];
    const float* b1  = (const float*)d_in[4];
    const float* W2  = (const float*)d_in[5];
    const float* b2  = (const float*)d_in[6];
    const float* Wr  = (const float*)d_in[7];
    const float* br  = (const float*)d_in[8];
    const int*   src = (const int*)d_in[9];
    const int*   dst = (const int*)d_in[10];

    float* out_v = (float*)d_out;                        // [NNODES,3,F]
    float* out_s = out_v + (size_t)NNODES * 3 * F;       // [NNODES,F]
    float* phi   = (float*)d_ws;                         // [NNODES,3F] = 76.8 MB

    (void)in_sizes; (void)n_in; (void)ws_size;

    hipMemsetAsync(d_out, 0, (size_t)out_size * sizeof(float), stream);

    node_mlp_kernel<<<NNODES / 16, 256, 0, stream>>>(s, W1, b1, W2, b2, phi);
    edge_kernel<<<(NEDGES + 7) / 8, 256, 0, stream>>>(v, rel, Wr, br, src, dst,
                                                      phi, out_v, out_s);
}